// RetinaNet_82532091560556
// MI455X (gfx1250) — compile-verified
//
#include <hip/hip_runtime.h>
#include <stdint.h>

#define NUM_CLASSES 80
#define TOPK        1000
#define NSEL        3000
#define SORTN       4096
#define NBINS       8192      // top 13 bits of monotonic float key
#define CAND_CAP    4096
#define TILES       188       // ceil(3000/16)
#define SUPW        94        // 3000 bits in u32 words
#define CONF_THRESH 0.05f
#define NMS_THRESH  0.6f
#define CTR_CLAMP   32.0f
#define SCALE_CLAMP 4.135166556742356f   // log(1000/16)
#define CLASS_OFFSET 4096.0f

typedef __attribute__((ext_vector_type(2))) float v2f;
typedef __attribute__((ext_vector_type(8))) float v8f;

struct LMeta { unsigned selCount, candCount, thrBin, cntAbove; };

// order-preserving float -> uint key (larger float => larger key)
__device__ __forceinline__ unsigned fkey(float f) {
  unsigned b = __float_as_uint(f);
  return (b & 0x80000000u) ? ~b : (b | 0x80000000u);
}
__device__ __forceinline__ float fkey_inv(unsigned u) {
  unsigned b = (u & 0x80000000u) ? (u ^ 0x80000000u) : ~u;
  return __uint_as_float(b);
}
__device__ __forceinline__ float score_of(float logit) {
  return sqrtf(1.0f / (1.0f + expf(-logit)));
}

// ---------------- init ----------------
__global__ void k_init(unsigned* hist, LMeta* meta) {
  int t = blockIdx.x * blockDim.x + threadIdx.x;
  if (t < 3 * NBINS) hist[t] = 0;
  if (t < 3) { meta[t].selCount = 0; meta[t].candCount = 0; meta[t].thrBin = 0; meta[t].cntAbove = 0; }
}

// ---------------- per-level histogram (streaming pass 1) ----------------
__global__ void k_hist(const float* __restrict__ cls, int n4, unsigned* __restrict__ hist) {
  __shared__ unsigned h[NBINS];
  for (int i = threadIdx.x; i < NBINS; i += blockDim.x) h[i] = 0;
  __syncthreads();
  const float4* p = (const float4*)cls;
  int stride = gridDim.x * blockDim.x;
  for (int i = blockIdx.x * blockDim.x + threadIdx.x; i < n4; i += stride) {
    __builtin_prefetch(p + i + stride, 0, 1);   // global_prefetch_b8
    float4 v = p[i];
    atomicAdd(&h[fkey(v.x) >> 19], 1u);
    atomicAdd(&h[fkey(v.y) >> 19], 1u);
    atomicAdd(&h[fkey(v.z) >> 19], 1u);
    atomicAdd(&h[fkey(v.w) >> 19], 1u);
  }
  __syncthreads();
  for (int i = threadIdx.x; i < NBINS; i += blockDim.x) {
    unsigned c = h[i];
    if (c) atomicAdd(&hist[i], c);
  }
}

// ---------------- find threshold bin ----------------
__global__ void k_selbin(const unsigned* __restrict__ hist, LMeta* meta) {
  if (threadIdx.x != 0) return;
  unsigned cum = 0, above = 0; int T = 0;
  for (int b = NBINS - 1; b >= 0; --b) {
    unsigned c = hist[b];
    if (cum + c >= TOPK) { T = b; above = cum; break; }
    cum += c;
  }
  meta->thrBin = (unsigned)T;
  meta->cntAbove = above;
}

// ---------------- compact (streaming pass 2) ----------------
__global__ void k_compact(const float* __restrict__ cls, int n4, LMeta* meta,
                          uint2* __restrict__ cand,
                          float* __restrict__ selScore, unsigned* __restrict__ selIdx) {
  unsigned T = meta->thrBin;
  const float4* p = (const float4*)cls;
  int stride = gridDim.x * blockDim.x;
  for (int i = blockIdx.x * blockDim.x + threadIdx.x; i < n4; i += stride) {
    __builtin_prefetch(p + i + stride, 0, 1);
    float4 v = p[i];
    float vals[4] = { v.x, v.y, v.z, v.w };
#pragma unroll
    for (int q = 0; q < 4; ++q) {
      unsigned key = fkey(vals[q]);
      unsigned bin = key >> 19;
      if (bin > T) {
        unsigned pos = atomicAdd(&meta->selCount, 1u);
        if (pos < TOPK) { selScore[pos] = score_of(vals[q]); selIdx[pos] = 4u * (unsigned)i + (unsigned)q; }
      } else if (bin == T) {
        unsigned pos = atomicAdd(&meta->candCount, 1u);
        if (pos < CAND_CAP) cand[pos] = make_uint2(key, 4u * (unsigned)i + (unsigned)q);
      }
    }
  }
}

// ---------------- in-LDS descending bitonic sort of SORTN u64 keys ----------------
__device__ void bitonic_desc(unsigned long long* keys) {
  for (unsigned k = 2; k <= SORTN; k <<= 1) {
    for (unsigned j = k >> 1; j > 0; j >>= 1) {
      __syncthreads();
      for (unsigned i = threadIdx.x; i < SORTN; i += blockDim.x) {
        unsigned ixj = i ^ j;
        if (ixj > i) {
          unsigned long long a = keys[i], b = keys[ixj];
          bool sw = ((i & k) == 0) ? (a < b) : (a > b);
          if (sw) { keys[i] = b; keys[ixj] = a; }
        }
      }
    }
  }
  __syncthreads();
}

// ---------------- resolve tie bin exactly + decode all TOPK of a level ----------------
__global__ void k_final_select(LMeta* meta, const uint2* __restrict__ cand,
                               float* __restrict__ selScore, unsigned* __restrict__ selIdx,
                               unsigned* __restrict__ selLabel, float* __restrict__ selBox,
                               unsigned* __restrict__ selValid,
                               const float* __restrict__ reg,
                               int W, float stride, float base) {
  __shared__ unsigned long long keys[SORTN];
  unsigned above = meta->cntAbove;
  unsigned cc = meta->candCount; if (cc > CAND_CAP) cc = CAND_CAP;
  int R = TOPK - (int)above; if (R < 0) R = 0;
  if ((unsigned)R > cc) R = (int)cc;

  for (int s = threadIdx.x; s < SORTN; s += blockDim.x) {
    unsigned long long kk = 0ull;
    if (s < (int)cc) {
      uint2 e = cand[s];
      kk = ((unsigned long long)e.x << 32) | (unsigned long long)(0xFFFFFFFFu - e.y); // ties: lower idx first
    }
    keys[s] = kk;
  }
  bitonic_desc(keys);

  for (int t = threadIdx.x; t < R; t += blockDim.x) {
    unsigned long long e = keys[t];
    unsigned key = (unsigned)(e >> 32);
    unsigned idx = 0xFFFFFFFFu - (unsigned)(e & 0xFFFFFFFFull);
    selScore[above + t] = score_of(fkey_inv(key));
    selIdx[above + t] = idx;
  }
  __syncthreads();

  const float arT[3] = { 0.5f, 1.0f, 2.0f };
  const float scT[3] = { 1.0f, 1.2599210498948732f, 1.5874010519681994f };
  for (int t = threadIdx.x; t < TOPK; t += blockDim.x) {
    unsigned idx = selIdx[t];
    unsigned a = idx / NUM_CLASSES;
    unsigned label = idx % NUM_CLASSES;
    unsigned p = a / 9, kk = a % 9;
    float cx = ((float)(p % (unsigned)W) + 0.5f) * stride;
    float cy = ((float)(p / (unsigned)W) + 0.5f) * stride;
    float ar = arT[kk / 3], sc = scT[kk % 3];
    float area = base * base * sc;
    float aw = sqrtf(ar * area), ah = sqrtf(area / ar);
    float4 r = ((const float4*)reg)[a];
    float ox = fminf(fmaxf(r.x * aw, -CTR_CLAMP), CTR_CLAMP);
    float oy = fminf(fmaxf(r.y * ah, -CTR_CLAMP), CTR_CLAMP);
    float ncx = cx + ox, ncy = cy + oy;
    float nw = aw * expf(fminf(r.z, SCALE_CLAMP));
    float nh = ah * expf(fminf(r.w, SCALE_CLAMP));
    selBox[t * 4 + 0] = ncx - 0.5f * nw;
    selBox[t * 4 + 1] = ncy - 0.5f * nh;
    selBox[t * 4 + 2] = ncx + 0.5f * nw;
    selBox[t * 4 + 3] = ncy + 0.5f * nh;
    selLabel[t] = label;
    selValid[t] = (selScore[t] > CONF_THRESH) ? 1u : 0u;
  }
}

// ---------------- global sort of 3000 candidates by score desc (stable) ----------------
__global__ void k_sort_all(const float* __restrict__ selScore, const unsigned* __restrict__ selLabel,
                           const float* __restrict__ selBox, const unsigned* __restrict__ selValid,
                           float* __restrict__ sScore, unsigned* __restrict__ sLabel,
                           float* __restrict__ sBox, unsigned* __restrict__ sValid,
                           float* __restrict__ sBo, float* __restrict__ sArea) {
  __shared__ unsigned long long keys[SORTN];
  for (int s = threadIdx.x; s < SORTN; s += blockDim.x) {
    unsigned long long kk = 0ull;
    if (s < NSEL)
      kk = ((unsigned long long)fkey(selScore[s]) << 32) | (unsigned long long)(0xFFFFFFFFu - (unsigned)s);
    keys[s] = kk;
  }
  bitonic_desc(keys);
  for (int i = threadIdx.x; i < NSEL; i += blockDim.x) {
    unsigned pos = 0xFFFFFFFFu - (unsigned)(keys[i] & 0xFFFFFFFFull);
    float s = selScore[pos];
    unsigned l = selLabel[pos];
    float b0 = selBox[pos * 4 + 0], b1 = selBox[pos * 4 + 1];
    float b2 = selBox[pos * 4 + 2], b3 = selBox[pos * 4 + 3];
    sScore[i] = s; sLabel[i] = l; sValid[i] = selValid[pos];
    sBox[i * 4 + 0] = b0; sBox[i * 4 + 1] = b1; sBox[i * 4 + 2] = b2; sBox[i * 4 + 3] = b3;
    float off = (float)l * CLASS_OFFSET;
    float x1 = b0 + off, y1 = b1 + off, x2 = b2 + off, y2 = b3 + off;
    sBo[i * 4 + 0] = x1; sBo[i * 4 + 1] = y1; sBo[i * 4 + 2] = x2; sBo[i * 4 + 3] = y2;
    sArea[i] = fmaxf(x2 - x1, 0.0f) * fmaxf(y2 - y1, 0.0f);
  }
}

// ---------------- IoU tile kernel: one wave per 16x16 tile, WMMA for area_i+area_j ----------------
__global__ void k_iou_tiles(const float* __restrict__ sBo, const float* __restrict__ sArea,
                            unsigned short* __restrict__ mask16) {
  int tj = blockIdx.x, ti = blockIdx.y;
  if (tj < ti) return;                        // only need j > i
  __shared__ float bi[16][4], bj[16][4], ai[16], aj[16];
  __shared__ unsigned rowbits[16];
  int lane = threadIdx.x;
  int i0 = ti * 16, j0 = tj * 16;
  if (lane < 16) {
    int i = i0 + lane;
    if (i < NSEL) {
      bi[lane][0] = sBo[i * 4 + 0]; bi[lane][1] = sBo[i * 4 + 1];
      bi[lane][2] = sBo[i * 4 + 2]; bi[lane][3] = sBo[i * 4 + 3];
      ai[lane] = sArea[i];
    } else { bi[lane][0] = bi[lane][1] = bi[lane][2] = bi[lane][3] = 0.f; ai[lane] = 0.f; }
    rowbits[lane] = 0u;
  } else {
    int n = lane - 16, j = j0 + n;
    if (j < NSEL) {
      bj[n][0] = sBo[j * 4 + 0]; bj[n][1] = sBo[j * 4 + 1];
      bj[n][2] = sBo[j * 4 + 2]; bj[n][3] = sBo[j * 4 + 3];
      aj[n] = sArea[j];
    } else { bj[n][0] = bj[n][1] = bj[n][2] = bj[n][3] = 0.f; aj[n] = 0.f; }
  }
  __syncthreads();

  int col = lane & 15;
  // D[m][n] = area_i[m] + area_j[n] via two chained v_wmma_f32_16x16x4_f32,
  // K-uniform operands make this independent of the A/B K-slot placement.
  float areai = ai[col];
  float areaj = aj[col];
  v2f A1 = { areai, areai };
  v2f B1 = { 0.25f, 0.25f };
  v8f acc = {};
  acc = __builtin_amdgcn_wmma_f32_16x16x4_f32(false, A1, false, B1, (short)0, acc, false, false);
  v2f A2 = { 0.25f, 0.25f };
  v2f B2 = { areaj, areaj };
  acc = __builtin_amdgcn_wmma_f32_16x16x4_f32(false, A2, false, B2, (short)0, acc, false, false);

  float jx1 = bj[col][0], jy1 = bj[col][1], jx2 = bj[col][2], jy2 = bj[col][3];
  int j = j0 + col;
#pragma unroll
  for (int v = 0; v < 8; ++v) {
    int m = v + ((lane >> 4) << 3);           // documented C/D layout
    int i = i0 + m;
    float ix1 = bi[m][0], iy1 = bi[m][1], ix2 = bi[m][2], iy2 = bi[m][3];
    float iw = fmaxf(fminf(ix2, jx2) - fmaxf(ix1, jx1), 0.0f);
    float ih = fmaxf(fminf(iy2, jy2) - fmaxf(iy1, jy1), 0.0f);
    float inter = iw * ih;
    float uni = fmaxf(acc[v] - inter, 1e-10f);
    float iou = inter / uni;
    bool pred = (iou > NMS_THRESH) && (j > i) && (j < NSEL) && (i < NSEL);
    if (pred) atomicOr(&rowbits[m], 1u << col);
  }
  __syncthreads();
  if (lane < 16) {
    int i = i0 + lane;
    if (i < NSEL) mask16[i * TILES + tj] = (unsigned short)rowbits[lane];
  }
}

// ---------------- greedy NMS replay over precomputed masks ----------------
__global__ void k_greedy(const unsigned* __restrict__ sValid,
                         const unsigned short* __restrict__ mask16,
                         unsigned* __restrict__ keep) {
  __shared__ unsigned sup[SUPW];
  __shared__ int kflag;
  for (int t = threadIdx.x; t < SUPW; t += blockDim.x) sup[t] = 0u;
  __syncthreads();
  for (int i = 0; i < NSEL; ++i) {
    if (threadIdx.x == 0) {
      int k = (sValid[i] != 0u) && !((sup[i >> 5] >> (i & 31)) & 1u);
      kflag = k;
      keep[i] = (unsigned)k;
    }
    __syncthreads();
    if (kflag) {
      int startTj = i >> 4;                    // tiles with tj >= ti contain all j > i
      for (int w = threadIdx.x; w < SUPW; w += blockDim.x) {
        unsigned bits = 0u;
        int tj0 = 2 * w, tj1 = 2 * w + 1;
        if (tj0 >= startTj) bits |= (unsigned)mask16[i * TILES + tj0];
        if (tj1 >= startTj) bits |= ((unsigned)mask16[i * TILES + tj1]) << 16;
        if (bits) sup[w] |= bits;
      }
    }
    __syncthreads();
  }
}

// ---------------- finalize outputs ----------------
__global__ void k_finalize(const float* __restrict__ sScore, const unsigned* __restrict__ sLabel,
                           const float* __restrict__ sBox, const unsigned* __restrict__ keep,
                           const int* __restrict__ img_h, const int* __restrict__ img_w,
                           float* __restrict__ out) {
  int i = blockIdx.x * blockDim.x + threadIdx.x;
  if (i >= NSEL) return;
  float Wf = (float)img_w[0], Hf = (float)img_h[0];
  unsigned kp = keep[i];
  float scl[4] = { Wf, Hf, Wf, Hf };
#pragma unroll
  for (int c = 0; c < 4; ++c) {
    float v = sBox[i * 4 + c] / scl[c];
    v = fminf(fmaxf(v, 0.0f), 1.0f);
    out[i * 4 + c] = kp ? v : 0.0f;
  }
  out[4 * NSEL + i] = kp ? sScore[i] : 0.0f;
  out[5 * NSEL + i] = (float)sLabel[i];
  out[6 * NSEL + i] = kp ? 1.0f : 0.0f;
}

extern "C" void kernel_launch(void* const* d_in, const int* in_sizes, int n_in,
                              void* d_out, int out_size, void* d_ws, size_t ws_size,
                              hipStream_t stream) {
  const float* cls[3] = { (const float*)d_in[0], (const float*)d_in[1], (const float*)d_in[2] };
  const float* reg[3] = { (const float*)d_in[3], (const float*)d_in[4], (const float*)d_in[5] };
  const int* img_h = (const int*)d_in[6];
  const int* img_w = (const int*)d_in[7];

  char* ws = (char*)d_ws;
  size_t off = 0;
  auto walloc = [&](size_t bytes) -> void* {
    void* p = (void*)(ws + off);
    off = (off + bytes + 255) & ~(size_t)255;
    return p;
  };
  unsigned*       hist     = (unsigned*)      walloc(3 * NBINS * sizeof(unsigned));
  LMeta*          meta     = (LMeta*)         walloc(3 * sizeof(LMeta));
  uint2*          cand     = (uint2*)         walloc(3 * CAND_CAP * sizeof(uint2));
  float*          selScore = (float*)         walloc(NSEL * sizeof(float));
  unsigned*       selIdx   = (unsigned*)      walloc(NSEL * sizeof(unsigned));
  unsigned*       selLabel = (unsigned*)      walloc(NSEL * sizeof(unsigned));
  float*          selBox   = (float*)         walloc(NSEL * 4 * sizeof(float));
  unsigned*       selValid = (unsigned*)      walloc(NSEL * sizeof(unsigned));
  float*          sScore   = (float*)         walloc(NSEL * sizeof(float));
  unsigned*       sLabel   = (unsigned*)      walloc(NSEL * sizeof(unsigned));
  float*          sBox     = (float*)         walloc(NSEL * 4 * sizeof(float));
  unsigned*       sValid   = (unsigned*)      walloc(NSEL * sizeof(unsigned));
  float*          sBo      = (float*)         walloc(NSEL * 4 * sizeof(float));
  float*          sArea    = (float*)         walloc(NSEL * sizeof(float));
  unsigned*       keep     = (unsigned*)      walloc(NSEL * sizeof(unsigned));
  unsigned short* mask16   = (unsigned short*)walloc((size_t)NSEL * TILES * sizeof(unsigned short));
  (void)ws_size; (void)in_sizes; (void)n_in; (void)out_size;

  k_init<<<(3 * NBINS + 255) / 256, 256, 0, stream>>>(hist, meta);

  const int   nEl[3]  = { 160 * 160 * 9 * NUM_CLASSES, 80 * 80 * 9 * NUM_CLASSES, 40 * 40 * 9 * NUM_CLASSES };
  const int   Wl[3]   = { 160, 80, 40 };
  const float strd[3] = { 8.0f, 16.0f, 32.0f };
  const float basl[3] = { 32.0f, 64.0f, 128.0f };

  for (int L = 0; L < 3; ++L) {
    int n4 = nEl[L] / 4;
    int blocks = (n4 + 255) / 256; if (blocks > 1024) blocks = 1024;
    k_hist<<<blocks, 256, 0, stream>>>(cls[L], n4, hist + L * NBINS);
    k_selbin<<<1, 32, 0, stream>>>(hist + L * NBINS, meta + L);
    k_compact<<<blocks, 256, 0, stream>>>(cls[L], n4, meta + L, cand + L * CAND_CAP,
                                          selScore + L * TOPK, selIdx + L * TOPK);
    k_final_select<<<1, 1024, 0, stream>>>(meta + L, cand + L * CAND_CAP,
                                           selScore + L * TOPK, selIdx + L * TOPK,
                                           selLabel + L * TOPK, selBox + L * TOPK * 4,
                                           selValid + L * TOPK, reg[L],
                                           Wl[L], strd[L], basl[L]);
  }

  k_sort_all<<<1, 1024, 0, stream>>>(selScore, selLabel, selBox, selValid,
                                     sScore, sLabel, sBox, sValid, sBo, sArea);
  k_iou_tiles<<<dim3(TILES, TILES), 32, 0, stream>>>(sBo, sArea, mask16);
  k_greedy<<<1, 128, 0, stream>>>(sValid, mask16, keep);
  k_finalize<<<(NSEL + 255) / 256, 256, 0, stream>>>(sScore, sLabel, sBox, keep, img_h, img_w,
                                                     (float*)d_out);
}